// QLayer_65481071404384
// MI455X (gfx1250) — compile-verified
//
#include <hip/hip_runtime.h>

// out[b,w] = prod_{k<=w} cos^2(x[b,k])   (closed form of the reference circuit;
// RZ params cancel, CNOT ladder => prefix-XOR => prefix product of Z-expectations).
//
// Prefix product done as exp( logm @ L ) where L is the 16x16 lower-triangular
// ones matrix, computed with 4x V_WMMA_F32_16X16X4_F32 per 16-row tile.

typedef float v2f __attribute__((ext_vector_type(2)));
typedef float v8f __attribute__((ext_vector_type(8)));

__global__ __launch_bounds__(32) void qlayer_prefix_wmma(const float* __restrict__ x,
                                                         float* __restrict__ out) {
  const int lane = threadIdx.x & 31;
  const int tile = blockIdx.x;
  const int nLow = lane & 15;   // column (N) for B/C/D; row (M) for A
  const int hi   = lane >> 4;   // half-wave select
  const int rowBase = tile * 16;

  // --- A matrix: logm[M][K], M = nLow for this lane.
  // ISA layout for 32-bit A 16x4: lanes 0-15 hold K = {4q, 4q+1},
  // lanes 16-31 hold K = {4q+2, 4q+3} (VGPR0 = even K, VGPR1 = odd K).
  const float* xrow = x + (size_t)(rowBase + nLow) * 16;

  float la[8];
#pragma unroll
  for (int q = 0; q < 4; ++q) {
#pragma unroll
    for (int j = 0; j < 2; ++j) {
      const int k = 4 * q + 2 * hi + j;
      const float c = cosf(xrow[k]);
      const float m = fmaxf(c * c, 1e-30f);  // avoid -inf (NaN vs. zeros in B)
      la[2 * q + j] = logf(m);
    }
  }

  // --- 4 chained WMMAs: acc[M][N] = sum_K logm[M][K] * (K <= N)
  v8f acc = {};
#pragma unroll
  for (int q = 0; q < 4; ++q) {
    const int k0 = 4 * q + 2 * hi;  // this lane's two K indices: k0, k0+1
    v2f A;
    A.x = la[2 * q];
    A.y = la[2 * q + 1];
    v2f B;  // lower-triangular ones: B[K][N] = (K <= N)
    B.x = (k0 <= nLow) ? 1.0f : 0.0f;
    B.y = (k0 + 1 <= nLow) ? 1.0f : 0.0f;
    acc = __builtin_amdgcn_wmma_f32_16x16x4_f32(
        /*neg_a=*/false, A, /*neg_b=*/false, B,
        /*c_mod=*/(short)0, acc, /*reuse_a=*/false, /*reuse_b=*/false);
  }

  // --- D layout: VGPR r holds (M = r + 8*hi, N = nLow). Exponentiate and store.
#pragma unroll
  for (int r = 0; r < 8; ++r) {
    const int M = r + 8 * hi;
    out[(size_t)(rowBase + M) * 16 + nLow] = expf(acc[r]);
  }
}

extern "C" void kernel_launch(void* const* d_in, const int* in_sizes, int n_in,
                              void* d_out, int out_size, void* d_ws, size_t ws_size,
                              hipStream_t stream) {
  const float* x = (const float*)d_in[0];
  // d_in[1] (params) provably does not affect the output: RZ phases vanish in
  // the probabilities and the CNOT ladder only permutes basis states.
  float* out = (float*)d_out;

  const int rows  = in_sizes[0] / 16;  // BATCH (= 512)
  const int tiles = rows / 16;         // 32 waves total

  qlayer_prefix_wmma<<<dim3(tiles), dim3(32), 0, stream>>>(x, out);
}